// AdaptiveGraphConv_74972949119134
// MI455X (gfx1250) — compile-verified
//
#include <hip/hip_runtime.h>
#include <hip/hip_bf16.h>

// ---------------------------------------------------------------------------
// AdaptiveGraphConv on MI455X (gfx1250, wave32, WMMA)
//
// Algebra:
//   out[n,o,t,u] = sum_{s,c,v} Weff[o, s*64+c] * x[n,c,t,v] * adaptive[n,s,v,u]
//   with Weff[o,k] = sum_s W_big[s*64+o, k]  (final sum over S folded into W)
// Per (n,t):
//   GEMM1: Ybig[o, s*25+v] = W_s(64x64) @ X_t(64x25)    (s = 0..2)
//   GEMM2: out_t(64x25)    = Ybig(64x75) @ adaptB(75x25)
// Both GEMMs run on V_WMMA_F32_16X16X4_F32 (exact fp32; problem is HBM-bound:
// ~125 MB min traffic -> ~5.4us at 23.3 TB/s vs only 8.2 GFLOP of math).
//
// B-operand tiles (Xs, Al) use a k-pair-interleaved LDS layout:
//   buf[(k>>1)*2*UP + col*2 + (k&1)]
// so each WMMA B-fragment {B[k][col], B[k+1][col]} is one aligned float2 ->
// a single ds_load_b64 into the operand pair, no repacking movs.
// ---------------------------------------------------------------------------

typedef __attribute__((ext_vector_type(2))) float v2f;
typedef __attribute__((ext_vector_type(8))) float v8f;

#define N_  32
#define C_  64
#define T_  300
#define V_  25
#define S_  3
#define E_  64
#define K_  192                 // S_*C_
#define J_  75                  // S_*V_
#define JP  80                  // J_ padded to multiple of 4 (K of GEMM2)
#define UP  32                  // V_ padded to multiple of 16 (N of both GEMMs)
#define TT  3                   // timesteps per block (300 % 3 == 0)

// interleaved B-tile index: row k (K-dim), column col
__device__ __forceinline__ int bidx(int k, int col) {
    return (k >> 1) * (2 * UP) + col * 2 + (k & 1);
}

// ---------------------------------------------------------------------------
// Kernel 0: fold S into W_big / b_big.
// ---------------------------------------------------------------------------
__global__ void prep_weights(const float* __restrict__ Wb,
                             const float* __restrict__ bb,
                             float* __restrict__ Weff,
                             float* __restrict__ beff) {
    int idx = blockIdx.x * blockDim.x + threadIdx.x;
    if (idx < C_ * K_) {
        int o = idx / K_, k = idx % K_;
        Weff[idx] = Wb[o * K_ + k] + Wb[(64 + o) * K_ + k] + Wb[(128 + o) * K_ + k];
    }
    if (idx < C_) {
        beff[idx] = bb[idx] + bb[64 + idx] + bb[128 + idx];
    }
}

// ---------------------------------------------------------------------------
// Kernel 1: temporal mean. One block per (n,c). Register partials over t,
// then LDS float atomics (ds_add_f32) to reduce across threads.
// ---------------------------------------------------------------------------
__global__ __launch_bounds__(128)
void temporal_mean(const float* __restrict__ x, float* __restrict__ xmean) {
    __shared__ float sums[V_];
    const int nc  = blockIdx.x;          // n*64 + c
    const int tid = threadIdx.x;
    if (tid < V_) sums[tid] = 0.0f;
    __syncthreads();

    const float* xp = x + (size_t)nc * (T_ * V_);
    float loc[V_];
#pragma unroll
    for (int v = 0; v < V_; ++v) loc[v] = 0.0f;
    for (int t = tid; t < T_; t += 128) {
#pragma unroll
        for (int v = 0; v < V_; ++v) loc[v] += xp[t * V_ + v];
    }
#pragma unroll
    for (int v = 0; v < V_; ++v) atomicAdd(&sums[v], loc[v]);
    __syncthreads();
    if (tid < V_) xmean[nc * V_ + tid] = sums[tid] * (1.0f / T_);
}

// ---------------------------------------------------------------------------
// Kernel 2: per-n adaptive adjacency.
//   theta/phi = W @ x_mean + b; sim = theta^T phi; softmax rows;
//   adaptB[n][ bidx(j = s*25+v, u) ] = A[s,v,u] + B[s,v,u] + softmax(sim)[v,u]
// Emitted directly in the k-pair-interleaved, zero-padded [80][32] layout.
// ---------------------------------------------------------------------------
__global__ __launch_bounds__(128)
void build_adaptive(const float* __restrict__ xmean,
                    const float* __restrict__ A,
                    const float* __restrict__ B,
                    const float* __restrict__ Wth, const float* __restrict__ bth,
                    const float* __restrict__ Wph, const float* __restrict__ bph,
                    float* __restrict__ adaptB) {
    __shared__ float xm[C_ * V_];
    __shared__ float th[E_ * V_];
    __shared__ float ph[E_ * V_];
    __shared__ float sim[V_ * V_];
    const int n = blockIdx.x, tid = threadIdx.x;

    for (int i = tid; i < C_ * V_; i += 128) xm[i] = xmean[n * C_ * V_ + i];
    __syncthreads();

    for (int i = tid; i < E_ * V_; i += 128) {
        int e = i / V_, v = i % V_;
        float st = 0.0f, sp = 0.0f;
        for (int c = 0; c < C_; ++c) {
            float xv = xm[c * V_ + v];
            st += Wth[e * C_ + c] * xv;
            sp += Wph[e * C_ + c] * xv;
        }
        th[i] = st + bth[e];
        ph[i] = sp + bph[e];
    }
    __syncthreads();

    for (int i = tid; i < V_ * V_; i += 128) {
        int v = i / V_, u = i % V_;
        float s = 0.0f;
        for (int e = 0; e < E_; ++e) s += th[e * V_ + v] * ph[e * V_ + u];
        sim[i] = s;
    }
    __syncthreads();

    if (tid < V_) {                       // softmax over u, one row per lane
        const int v = tid;
        float m = sim[v * V_];
        for (int u = 1; u < V_; ++u) m = fmaxf(m, sim[v * V_ + u]);
        float ssum = 0.0f;
        for (int u = 0; u < V_; ++u) {
            float e = __expf(sim[v * V_ + u] - m);
            sim[v * V_ + u] = e;
            ssum += e;
        }
        float inv = 1.0f / ssum;
        for (int u = 0; u < V_; ++u) sim[v * V_ + u] *= inv;
    }
    __syncthreads();

    float* apn = adaptB + (size_t)n * JP * UP;
    for (int i = tid; i < JP * UP; i += 128) {
        int j = i / UP, u = i % UP;
        float val = 0.0f;
        if (j < J_ && u < V_) {
            int s = j / V_, v = j % V_;
            val = A[(s * V_ + v) * V_ + u] + B[(s * V_ + v) * V_ + u] + sim[v * V_ + u];
        }
        apn[bidx(j, u)] = val;
    }
}

// ---------------------------------------------------------------------------
// Kernel 3: fused main pass. One block per (n, 3-timestep tile), 4 waves;
// wave w owns the 16-row o-tile [16w, 16w+16). Dual WMMA GEMMs through LDS.
//
// V_WMMA_F32_16X16X4_F32 lane layouts (ISA 7.12.2):
//   A 16x4 : lane = m (0..15) | m+16 ; VGPR0/1 = K {0,1} (lanes 0-15) or {2,3}
//   B 4x16 : lane = n (0..15) | n+16 ; VGPR0/1 = K {0,1} (lanes 0-15) or {2,3}
//   C/D    : VGPR r -> rows r (lanes 0-15) / r+8 (lanes 16-31), col = lane&15
// Fragment k is even (k = 4*kk + kh, kh in {0,2}) -> every A/B fragment is an
// aligned float2: global_load_b64 / ds_load_b64 straight into operand pairs.
// ---------------------------------------------------------------------------
__global__ __launch_bounds__(128)
void fused_graph_conv(const float* __restrict__ x,
                      const float* __restrict__ Weff,
                      const float* __restrict__ beff,
                      const float* __restrict__ adaptB,
                      float* __restrict__ out) {
    __shared__ float Xs[C_ * UP];    // X_t   interleaved [c-pair][u][2]   8 KB
    __shared__ float Yl[C_ * JP];    // Ybig  row-major   [o=64][j pad 80] 20 KB
    __shared__ float Al[JP * UP];    // adaptB interleaved [j-pair][u][2]  10 KB
    __shared__ float bl[C_];         // beff                               256 B

    const int bid = blockIdx.x;
    const int n  = bid / (T_ / TT);
    const int t0 = (bid % (T_ / TT)) * TT;
    const int tid = threadIdx.x;

    // Stage adaptB[n] (already interleaved+padded) and beff once per block.
    const float* apn = adaptB + (size_t)n * JP * UP;
    for (int i = tid; i < JP * UP; i += 128) Al[i] = apn[i];
    if (tid < C_) bl[tid] = beff[tid];
    // Zero Ybig's K-padding columns (j = 75..79); GEMM1 never writes them.
    for (int i = tid; i < C_ * (JP - J_); i += 128) {
        int row = i / (JP - J_), col = J_ + i % (JP - J_);
        Yl[row * JP + col] = 0.0f;
    }

    const int lane = tid & 31;
    const int wave = tid >> 5;
    const int mloc = lane & 15;          // M (A) / N (B,D) position
    const int kh   = (lane >> 4) * 2;    // K sub-offset: 0 or 2
    const int orow = wave * 16 + mloc;   // this lane's A-matrix row (o)

    const float* xbase = x + (size_t)(n * C_) * (T_ * V_);

#pragma unroll 1
    for (int it = 0; it < TT; ++it) {
        const int t = t0 + it;
        // Stage X_t into interleaved layout, zero pad v>=25.
        const float* xp = xbase + (size_t)t * V_;
        for (int i = tid; i < C_ * UP; i += 128) {
            int c = i >> 5, v = i & 31;
            Xs[bidx(c, v)] = (v < V_) ? xp[(size_t)c * (T_ * V_) + v] : 0.0f;
        }
        // Prefetch next timestep's rows (2 cachelines per 100B row) so the
        // strided fetch hides behind this step's WMMA work.
        if (it + 1 < TT) {
            const float* xn = xbase + (size_t)(t + 1) * V_;
            int c = tid >> 1, half = tid & 1;
            __builtin_prefetch(xn + (size_t)c * (T_ * V_) + half * 16, 0, 1);
        }
        __syncthreads();

        // ---- GEMM1: Ybig[o, s*25+v] = W_s @ X_t  (K = 64, 16 k-steps) ----
#pragma unroll
        for (int s = 0; s < S_; ++s) {
            const float* wrow = Weff + (size_t)orow * K_ + s * C_;
#pragma unroll
            for (int nt = 0; nt < 2; ++nt) {
                v8f acc = {0.f, 0.f, 0.f, 0.f, 0.f, 0.f, 0.f, 0.f};
#pragma unroll
                for (int kk = 0; kk < C_ / 4; ++kk) {
                    const int k = kk * 4 + kh;
                    v2f a = *(const v2f*)&wrow[k];                       // b64
                    v2f b = *(const v2f*)&Xs[bidx(k, nt * 16 + mloc)];   // b64
                    acc = __builtin_amdgcn_wmma_f32_16x16x4_f32(
                        false, a, false, b, (short)0, acc, false, false);
                }
                const int col = nt * 16 + mloc;
                if (col < V_) {
                    const int jcol = s * V_ + col;
#pragma unroll
                    for (int r = 0; r < 8; ++r) {
                        int row = (lane < 16) ? r : (r + 8);
                        Yl[(wave * 16 + row) * JP + jcol] = acc[r];
                    }
                }
            }
        }
        __syncthreads();

        // ---- GEMM2: out_t[o, u] = Ybig @ adaptB  (K = 80, 20 k-steps) ----
        // (Yl rows are produced and consumed by the same wave, so the next
        //  iteration's Xs restaging only needs the barrier above.)
#pragma unroll
        for (int nt = 0; nt < 2; ++nt) {
            v8f acc = {0.f, 0.f, 0.f, 0.f, 0.f, 0.f, 0.f, 0.f};
#pragma unroll
            for (int kk = 0; kk < JP / 4; ++kk) {
                const int k = kk * 4 + kh;
                v2f a = *(const v2f*)&Yl[orow * JP + k];                 // b64
                v2f b = *(const v2f*)&Al[bidx(k, nt * 16 + mloc)];       // b64
                acc = __builtin_amdgcn_wmma_f32_16x16x4_f32(
                    false, a, false, b, (short)0, acc, false, false);
            }
            const int u = nt * 16 + mloc;
            if (u < V_) {
#pragma unroll
                for (int r = 0; r < 8; ++r) {
                    int row = (lane < 16) ? r : (r + 8);
                    int o = wave * 16 + row;
                    out[(((size_t)n * C_ + o) * T_ + t) * V_ + u] = acc[r] + bl[o];
                }
            }
        }
    }
}

// ---------------------------------------------------------------------------
// Launch. ws layout (floats):
//   Weff   : 64*192        = 12288
//   beff   : 64
//   xmean  : 32*64*25      = 51200
//   adaptB : 32*80*32      = 81920   (k-pair interleaved)
// Total 145,472 floats (~582 KB).
// ---------------------------------------------------------------------------
extern "C" void kernel_launch(void* const* d_in, const int* in_sizes, int n_in,
                              void* d_out, int out_size, void* d_ws, size_t ws_size,
                              hipStream_t stream) {
    const float* x   = (const float*)d_in[0];
    const float* A   = (const float*)d_in[1];
    const float* B   = (const float*)d_in[2];
    const float* Wth = (const float*)d_in[3];
    const float* bth = (const float*)d_in[4];
    const float* Wph = (const float*)d_in[5];
    const float* bph = (const float*)d_in[6];
    const float* Wb  = (const float*)d_in[7];
    const float* bb  = (const float*)d_in[8];
    float* out = (float*)d_out;

    float* ws     = (float*)d_ws;
    float* Weff   = ws;                          // 12288
    float* beff   = Weff + C_ * K_;              // 64
    float* xmean  = beff + C_;                   // 51200
    float* adaptB = xmean + N_ * C_ * V_;        // 81920

    prep_weights<<<(C_ * K_ + 255) / 256, 256, 0, stream>>>(Wb, bb, Weff, beff);
    temporal_mean<<<N_ * C_, 128, 0, stream>>>(x, xmean);
    build_adaptive<<<N_, 128, 0, stream>>>(xmean, A, B, Wth, bth, Wph, bph, adaptB);
    fused_graph_conv<<<N_ * (T_ / TT), 128, 0, stream>>>(x, Weff, beff, adaptB, out);
}